// RPN_65695819759988
// MI455X (gfx1250) — compile-verified
//
#include <hip/hip_runtime.h>
#include <stdint.h>

// ---------------------------------------------------------------------------
// RPN pre-NMS top-k for MI455X (gfx1250).
// One 1024-thread workgroup per (level, batch) row:
//   1) 3-pass LDS radix select (11/11/10 bits of sign-flipped float bits)
//      to find the exact k-th-largest threshold,
//   2) compaction of >T (and ==T ties) into a 2048-entry LDS candidate list,
//   3) in-LDS bitonic sort of 64-bit keys (ord_bits<<32 | ~nhwc_index) so
//      ties order by ascending index like lax.top_k,
//   4) write values + (index + offset - 1).
// All global->LDS streaming uses the Tensor Data Mover (tensor_load_to_lds,
// double buffered, TENSORcnt-synchronized) — the CDNA5 async data path.
// ---------------------------------------------------------------------------

#if defined(__HIP_DEVICE_COMPILE__) && defined(__gfx1250__)
#define USE_TDM 1
#else
#define USE_TDM 0
#endif

#define TPB   1024
#define K_TOP 1000
#define TILE  4096          // staged elements per tile (16 KB)
#define NROWS (5 * 64)

typedef unsigned int v4u __attribute__((ext_vector_type(4)));
typedef unsigned int v8u __attribute__((ext_vector_type(8)));

__device__ __forceinline__ uint32_t f2ord(uint32_t f) {
  // monotonic (ascending) transform of float bits
  return f ^ ((f & 0x80000000u) ? 0xFFFFFFFFu : 0x80000000u);
}
__device__ __forceinline__ uint32_t ord2f(uint32_t u) {
  return u ^ ((u & 0x80000000u) ? 0x80000000u : 0xFFFFFFFFu);
}

#if USE_TDM
// Build a TDM descriptor (CDNA5 ISA) for a 1-D tile of
// `nelem` 4-byte elements and issue tensor_load_to_lds via inline asm
// (portable across ROCm7.2 / amdgpu-toolchain per CDNA5_HIP.md).
__device__ __forceinline__ void tdm_issue(const uint32_t* gsrc, int nelem,
                                          uint32_t* lds_dst) {
  unsigned long long ga = (unsigned long long)gsrc;
  unsigned lds = (unsigned)(unsigned long long)(void*)lds_dst;  // LDS byte addr
  unsigned n = (unsigned)nelem;

  v4u g0;
  g0.x = 1u;                                   // count=1, user mode, no gather
  g0.y = lds;                                  // lds_addr
  g0.z = (unsigned)ga;                         // global_addr[31:0]
  g0.w = (unsigned)((ga >> 32) & 0x01FFFFFFu)  // global_addr[56:32]
         | 0x80000000u;                        // type=2 ("image")

  v8u g1;
  g1.s0 = 2u << 16;                            // data_size=2 -> 4 bytes
  g1.s1 = (n & 0xFFFFu) << 16;                 // tensor_dim0[15:0] (abar=0)
  g1.s2 = (n >> 16) | (1u << 16);              // tensor_dim0[31:16], tensor_dim1=1
  g1.s3 = n << 16;                             // tensor_dim1[31:16]=0, tile_dim0=n
  g1.s4 = 1u;                                  // tile_dim1=1, tile_dim2=0
  g1.s5 = n;                                   // tensor_dim0_stride[31:0]
  g1.s6 = (n & 0xFFFFu) << 16;                 // d0s[47:32]=0, d1s[15:0]
  g1.s7 = n >> 16;                             // tensor_dim1_stride[47:16]

  asm volatile("tensor_load_to_lds %0, %1" :: "s"(g0), "s"(g1) : "memory");
}
#endif

// Stream one row through LDS, delivering (index, float_bits) to `f`.
template <typename F>
__device__ __forceinline__ void stream_row(const uint32_t* __restrict__ src,
                                           int n, uint32_t* stage, F f) {
  const int tid = (int)threadIdx.x;
#if USE_TDM
  const int ntiles = (n + TILE - 1) / TILE;
  if (tid < 32) {                               // wave 0 issues the DMA
    int nb0 = n < TILE ? n : TILE;
    tdm_issue(src, nb0, stage);
  }
  for (int t = 0; t < ntiles; ++t) {
    uint32_t* buf = stage + (t & 1) * TILE;
    if (tid < 32) {
      if (t + 1 < ntiles) {                     // prefetch next tile
        int nb = n - (t + 1) * TILE;
        if (nb > TILE) nb = TILE;
        tdm_issue(src + (t + 1) * TILE, nb, stage + ((t + 1) & 1) * TILE);
        __builtin_amdgcn_s_wait_tensorcnt(1);   // tile t landed
      } else {
        __builtin_amdgcn_s_wait_tensorcnt(0);
      }
    }
    __syncthreads();                            // publish LDS to all waves
    const int base = t * TILE;
    int cnt = n - base;
    if (cnt > TILE) cnt = TILE;
    for (int q = tid; q < cnt; q += TPB) f(base + q, buf[q]);
    __syncthreads();                            // done reading before reuse
  }
#else
  (void)stage;
  for (int i = tid; i < n; i += TPB) f(i, src[i]);
#endif
}

__global__ __launch_bounds__(TPB) void rpn_topk_kernel(
    const float* __restrict__ in0, const float* __restrict__ in1,
    const float* __restrict__ in2, const float* __restrict__ in3,
    const float* __restrict__ in4, float* __restrict__ out_val,
    int* __restrict__ out_idx) {
  __shared__ uint32_t stage[2 * TILE];          // 32 KB TDM double buffer
  __shared__ uint32_t hist[2048];               // 8 KB radix histogram
  __shared__ unsigned long long cand[2048];     // 16 KB candidate keys
  __shared__ uint32_t s_sel[4];                 // prefix, krem, cntG, cntE

  const int tid = (int)threadIdx.x;
  const int lev = (int)blockIdx.x >> 6;
  const int bat = (int)blockIdx.x & 63;

  int N, HW, OFF;
  const float* srcf;
  switch (lev) {
    case 0:  N = 201600; HW = 67200; OFF = 0;      srcf = in0; break;
    case 1:  N = 50400;  HW = 16800; OFF = 201600; srcf = in1; break;
    case 2:  N = 12600;  HW = 4200;  OFF = 252000; srcf = in2; break;
    case 3:  N = 3150;   HW = 1050;  OFF = 264600; srcf = in3; break;
    default: N = 1152;   HW = 384;   OFF = 267750; srcf = in4; break;
  }
  const uint32_t* src = (const uint32_t*)srcf + (size_t)bat * (size_t)N;

  if (tid == 0) { s_sel[0] = 0u; s_sel[1] = K_TOP; }

  // ---- radix pass 0: bits [31:21] -----------------------------------------
  for (int i = tid; i < 2048; i += TPB) hist[i] = 0u;
  __syncthreads();
  stream_row(src, N, stage, [&](int i, uint32_t fb) {
    (void)i;
    atomicAdd(&hist[f2ord(fb) >> 21], 1u);
  });
  __syncthreads();
  if (tid == 0) {
    uint32_t krem = s_sel[1], cum = 0u, bsel = 0u;
    for (int bin = 2047; bin >= 0; --bin) {
      uint32_t h = hist[bin];
      if (cum + h >= krem) { bsel = (uint32_t)bin; break; }
      cum += h;
    }
    s_sel[0] = bsel;
    s_sel[1] = krem - cum;
  }
  __syncthreads();

  // ---- radix pass 1: bits [20:10] -----------------------------------------
  {
    const uint32_t pfx = s_sel[0];
    for (int i = tid; i < 2048; i += TPB) hist[i] = 0u;
    __syncthreads();
    stream_row(src, N, stage, [&](int i, uint32_t fb) {
      (void)i;
      uint32_t u = f2ord(fb);
      if ((u >> 21) == pfx) atomicAdd(&hist[(u >> 10) & 0x7FFu], 1u);
    });
    __syncthreads();
    if (tid == 0) {
      uint32_t krem = s_sel[1], cum = 0u, bsel = 0u;
      for (int bin = 2047; bin >= 0; --bin) {
        uint32_t h = hist[bin];
        if (cum + h >= krem) { bsel = (uint32_t)bin; break; }
        cum += h;
      }
      s_sel[0] = (s_sel[0] << 11) | bsel;
      s_sel[1] = krem - cum;
    }
    __syncthreads();
  }

  // ---- radix pass 2: bits [9:0] -------------------------------------------
  {
    const uint32_t pfx = s_sel[0];              // 22-bit prefix
    for (int i = tid; i < 1024; i += TPB) hist[i] = 0u;
    __syncthreads();
    stream_row(src, N, stage, [&](int i, uint32_t fb) {
      (void)i;
      uint32_t u = f2ord(fb);
      if ((u >> 10) == pfx) atomicAdd(&hist[u & 0x3FFu], 1u);
    });
    __syncthreads();
    if (tid == 0) {
      uint32_t krem = s_sel[1], cum = 0u, bsel = 0u;
      for (int bin = 1023; bin >= 0; --bin) {
        uint32_t h = hist[bin];
        if (cum + h >= krem) { bsel = (uint32_t)bin; break; }
        cum += h;
      }
      s_sel[0] = (s_sel[0] << 10) | bsel;       // full 32-bit threshold
      s_sel[1] = krem - cum;
    }
    __syncthreads();
  }

  // ---- compaction: >T into [0..), ==T ties into [1024..) ------------------
  const uint32_t T = s_sel[0];
  if (tid == 0) { s_sel[2] = 0u; s_sel[3] = 0u; }
  for (int i = tid; i < 2048; i += TPB) cand[i] = 0ull;   // pad keys
  __syncthreads();
  stream_row(src, N, stage, [&](int i, uint32_t fb) {
    uint32_t u = f2ord(fb);
    if (u < T) return;
    int c = (i >= HW) + (i >= 2 * HW);                    // channel 0..2
    uint32_t nhwc = (uint32_t)((i - c * HW) * 3 + c);     // NHWC flat index
    unsigned long long key =
        ((unsigned long long)u << 32) | (uint32_t)(~nhwc);
    if (u > T) {
      uint32_t p = atomicAdd(&s_sel[2], 1u);              // p < K_TOP
      cand[p] = key;
    } else {
      uint32_t p = atomicAdd(&s_sel[3], 1u);
      if (p < 1024u) cand[1024u + p] = key;
    }
  });
  __syncthreads();

  // ---- bitonic sort, descending (ties -> smaller index first) -------------
  for (int kk = 2; kk <= 2048; kk <<= 1) {
    for (int j = kk >> 1; j > 0; j >>= 1) {
      __syncthreads();
      for (int i = tid; i < 2048; i += TPB) {
        int l = i ^ j;
        if (l > i) {
          unsigned long long a = cand[i];
          unsigned long long c2 = cand[l];
          bool up = (i & kk) == 0;
          bool sw = up ? (a < c2) : (a > c2);
          if (sw) { cand[i] = c2; cand[l] = a; }
        }
      }
    }
  }
  __syncthreads();

  // ---- emit top K ---------------------------------------------------------
  const size_t row = (size_t)((lev << 6) + bat) * (size_t)K_TOP;
  for (int jj = tid; jj < K_TOP; jj += TPB) {
    unsigned long long key = cand[jj];
    uint32_t u = (uint32_t)(key >> 32);
    uint32_t nhwc = ~((uint32_t)key);
    out_val[row + jj] = __uint_as_float(ord2f(u));
    out_idx[row + jj] = (int)nhwc + OFF - 1;              // faithful "-1" quirk
  }
}

extern "C" void kernel_launch(void* const* d_in, const int* in_sizes, int n_in,
                              void* d_out, int out_size, void* d_ws,
                              size_t ws_size, hipStream_t stream) {
  (void)in_sizes; (void)n_in; (void)out_size; (void)d_ws; (void)ws_size;
  const float* in0 = (const float*)d_in[0];
  const float* in1 = (const float*)d_in[1];
  const float* in2 = (const float*)d_in[2];
  const float* in3 = (const float*)d_in[3];
  const float* in4 = (const float*)d_in[4];
  float* out_val = (float*)d_out;
  int* out_idx = (int*)((float*)d_out + (size_t)NROWS * K_TOP);
  rpn_topk_kernel<<<dim3(NROWS), dim3(TPB), 0, stream>>>(
      in0, in1, in2, in3, in4, out_val, out_idx);
}